// MixCrossAttention_88046829568164
// MI455X (gfx1250) — compile-verified
//
#include <hip/hip_runtime.h>
#include <math.h>

// Sizes (fixed by the reference)
#define BB 4
#define HH 8
#define LL 1024
#define EE 64
#define NTOP 35
#define JPAD 48     // padded rows per attention path (3 tiles of 16)
#define JTOT 96     // both paths
#define NMODE 64

typedef __attribute__((ext_vector_type(2))) float v2f;
typedef __attribute__((ext_vector_type(8))) float v8f;

// ---- CDNA5 WMMA f32 16x16x4 wrapper -------------------------------------
// Layout (per ISA 7.12.2, wave32): lane = (n | hi<<4), n = lane&15, hi = lane>>4
//   A (16x4, MxK): component c -> element A[m = n][k = 2*hi + c]
//   B (4x16, KxN): component c -> element B[k = 2*hi + c][n]
//   C (16x16):     c[v]        -> element C[m = v + 8*hi][n]
__device__ __forceinline__ v8f wmma_f32(v2f a, v2f b, v8f c) {
  return __builtin_amdgcn_wmma_f32_16x16x4_f32(false, a, false, b, (short)0, c,
                                               false, false);
}

// input layout helpers: tensors (B,L,H,E) row-major
__device__ __forceinline__ size_t qidx(int b, int l, int h, int e) {
  return ((((size_t)b * LL + l) * HH + h) * EE) + e;
}

// =========================================================================
// Kernel 0: build forward DFT (1024 x 128: [cos | -sin]) and inverse DFT
// (128 x 1024) matrices.  IDFT folds: irfft 1/N, x2 for modes>=1, the
// /(IN_CH*OUT_CH) = /512^2, and the final *0.5.
// =========================================================================
__global__ void k_build_dft(float* __restrict__ dftF, float* __restrict__ dftI) {
  int t = blockIdx.x * blockDim.x + threadIdx.x;
  if (t >= LL * NMODE) return;
  int l = t >> 6, x = t & 63;
  const float w = 6.28318530717958647692f / 1024.0f;
  float ph = (float)((l * x) & 1023) * w;   // exact periodic reduction
  float s, c;
  __sincosf(ph, &s, &c);
  dftF[l * 128 + x] = c;
  dftF[l * 128 + 64 + x] = -s;
  float coef = 0.5f * ((x == 0) ? 1.0f : 2.0f) / 1024.0f / (512.0f * 512.0f);
  dftI[x * 1024 + l] = coef * c;
  dftI[(64 + x) * 1024 + l] = -coef * s;
}

// =========================================================================
// 1024-point radix-2 complex FFT in LDS, 256 threads.
// =========================================================================
__device__ void fft1024(float* re, float* im, int inverse) {
  int t = threadIdx.x;
  // bit-reversal
  for (int q = 0; q < 4; ++q) {
    int i = t + q * 256;
    int r = (int)(__brev((unsigned)i) >> 22);
    if (r > i) {
      float a = re[i]; re[i] = re[r]; re[r] = a;
      float b = im[i]; im[i] = im[r]; im[r] = b;
    }
  }
  __syncthreads();
  float sign = inverse ? 6.28318530717958647692f : -6.28318530717958647692f;
  for (int s = 1; s <= 10; ++s) {
    int half = 1 << (s - 1);
    for (int q = 0; q < 2; ++q) {
      int j = t + q * 256;                 // 512 butterflies / stage
      int grp = j >> (s - 1);
      int pos = j & (half - 1);
      int i0 = (grp << s) + pos;
      int i1 = i0 + half;
      float ang = sign * (float)pos / (float)(1 << s);
      float sn, cs;
      __sincosf(ang, &sn, &cs);
      float vr = re[i1] * cs - im[i1] * sn;
      float vi = re[i1] * sn + im[i1] * cs;
      float ur = re[i0], ui = im[i0];
      re[i0] = ur + vr; im[i0] = ui + vi;
      re[i1] = ur - vr; im[i1] = ui - vi;
    }
    __syncthreads();
  }
}

// =========================================================================
// Kernel 1: circular cross-correlation amplitude (corr=True ranking).
// One block per (b,h,e): amp_t[b,h,tau] += (irfft(Q * conj(K))[tau])^2
// =========================================================================
__global__ __launch_bounds__(256) void k_corr_amp(const float* __restrict__ qg,
                                                  const float* __restrict__ kg,
                                                  float* __restrict__ amp) {
  __shared__ float ar[1024], ai[1024], br[1024], bi[1024];
  int bh = blockIdx.x >> 6;
  int e = blockIdx.x & 63;
  int b = bh >> 3, h = bh & 7;
  int t = threadIdx.x;
  size_t base = qidx(b, 0, h, e);
  for (int q = 0; q < 4; ++q) {
    int l = t + q * 256;
    ar[l] = qg[base + (size_t)l * 512]; ai[l] = 0.f;
    br[l] = kg[base + (size_t)l * 512]; bi[l] = 0.f;
  }
  __syncthreads();
  fft1024(ar, ai, 0);
  fft1024(br, bi, 0);
  for (int q = 0; q < 4; ++q) {
    int l = t + q * 256;
    float qr = ar[l], qi = ai[l], kr = br[l], ki = bi[l];
    ar[l] = qr * kr + qi * ki;     // Q * conj(K)
    ai[l] = qi * kr - qr * ki;
  }
  __syncthreads();
  fft1024(ar, ai, 1);
  const float inv = 1.0f / 1024.0f;
  for (int q = 0; q < 4; ++q) {
    int l = t + q * 256;
    float v = ar[l] * inv;
    atomicAdd(&amp[bh * 1024 + l], v * v);
  }
}

// =========================================================================
// Kernel 2: tf-path amplitude = ||tf_q[b,l,h,:]||^2 (monotone with norm).
// =========================================================================
__global__ void k_tf_amp(const float* __restrict__ tq, float* __restrict__ amp) {
  int t = blockIdx.x * blockDim.x + threadIdx.x;
  if (t >= BB * HH * LL) return;
  int bh = t >> 10, l = t & 1023;
  int b = bh >> 3, h = bh & 7;
  const float4* p = (const float4*)(tq + qidx(b, l, h, 0));
  float s = 0.f;
#pragma unroll
  for (int i = 0; i < 16; ++i) {
    float4 v = p[i];
    s += v.x * v.x + v.y * v.y + v.z * v.z + v.w * v.w;
  }
  amp[bh * 1024 + l] = s;
}

// =========================================================================
// Kernel 3: top-35 via iterative argmax (ties -> smaller index, like top_k).
// grid: 64 blocks = 2 paths * 32 (b,h)
// =========================================================================
__global__ __launch_bounds__(256) void k_topk(const float* __restrict__ ampT,
                                              const float* __restrict__ ampTF,
                                              int* __restrict__ idxT,
                                              int* __restrict__ idxTF) {
  __shared__ float val[1024];
  __shared__ float rv[256];
  __shared__ int ri[256];
  int path = blockIdx.x >> 5;
  int bh = blockIdx.x & 31;
  const float* amp = path ? ampTF : ampT;
  int* idx = path ? idxTF : idxT;
  int t = threadIdx.x;
  for (int q = 0; q < 4; ++q) val[t + q * 256] = amp[bh * 1024 + t + q * 256];
  __syncthreads();
  for (int r = 0; r < NTOP; ++r) {
    float mx = -1e30f; int mi = 1024;
    for (int q = 0; q < 4; ++q) {
      int i = t + q * 256;
      float v = val[i];
      if (v > mx || (v == mx && i < mi)) { mx = v; mi = i; }
    }
    rv[t] = mx; ri[t] = mi;
    __syncthreads();
    for (int s2 = 128; s2 > 0; s2 >>= 1) {
      if (t < s2) {
        float ov = rv[t + s2]; int oi = ri[t + s2];
        if (ov > rv[t] || (ov == rv[t] && oi < ri[t])) { rv[t] = ov; ri[t] = oi; }
      }
      __syncthreads();
    }
    if (t == 0) { idx[bh * NTOP + r] = ri[0]; val[ri[0]] = -1e30f; }
    __syncthreads();
  }
}

// =========================================================================
// Kernel 4: attention. One block per (b,h), 8 waves, each wave owns s-tiles.
// Scores: WMMA  C[j, s] = sum_d Qsel[j,d] * K[s,d], both paths (j padded 48)
// Softmax over j (masked), fuse weights, then WMMA  out[s,d] = P^T V (K=96).
// Writes d_out = 0.5*attn (full overwrite -> replay-safe).
// =========================================================================
__global__ __launch_bounds__(256) void k_attn(
    const float* __restrict__ qg, const float* __restrict__ tqg,
    const float* __restrict__ kg, const float* __restrict__ vg,
    const float* __restrict__ wfuse, const int* __restrict__ idxT,
    const int* __restrict__ idxTF, float* __restrict__ out) {
  __shared__ int jdx[JTOT];
  __shared__ float Plds[8 * JTOT * 16];   // 48 KB, per-wave slabs
  int b = blockIdx.x >> 3, h = blockIdx.x & 7;
  int t = threadIdx.x;
  if (t < JTOT) {
    int j = t, v = 0;
    if (j < NTOP) v = idxT[(b * 8 + h) * NTOP + j];
    else if (j >= JPAD && j < JPAD + NTOP) v = idxTF[(b * 8 + h) * NTOP + (j - JPAD)];
    jdx[j] = v;
  }
  __syncthreads();
  int wid = t >> 5, lane = t & 31, n = lane & 15, hi = lane >> 4;
  float* Pw = &Plds[wid * JTOT * 16];
  size_t bhq = qidx(b, 0, h, 0);
  const float SC = 0.04419417382415922f;  // 1/sqrt(512)

  for (int st = wid; st < 64; st += 8) {
    int s0 = st * 16;
    // ---- score GEMMs:  C[m=j, n=s], K = d (64) ----
    v8f ct[3] = {}, cf[3] = {};
    for (int kk = 0; kk < 16; ++kk) {
      int d0 = kk * 4 + hi * 2;
      size_t kb = bhq + (size_t)(s0 + n) * 512;
      v2f bb; bb.x = kg[kb + d0]; bb.y = kg[kb + d0 + 1];
#pragma unroll
      for (int mt = 0; mt < 3; ++mt) {
        int j = mt * 16 + n;
        size_t qa = bhq + (size_t)jdx[j] * 512;
        v2f a; a.x = qg[qa + d0]; a.y = qg[qa + d0 + 1];
        ct[mt] = wmma_f32(a, bb, ct[mt]);
        size_t qa2 = bhq + (size_t)jdx[JPAD + j] * 512;
        v2f a2; a2.x = tqg[qa2 + d0]; a2.y = tqg[qa2 + d0 + 1];
        cf[mt] = wmma_f32(a2, bb, cf[mt]);
      }
    }
    // ---- fuse weight softmax (per key position s) ----
    int s = s0 + n;
    float wf0 = wfuse[(h * 1024 + s) * 2 + 0];
    float wf1 = wfuse[(h * 1024 + s) * 2 + 1];
    float wm = fmaxf(wf0, wf1);
    float e0 = __expf(wf0 - wm), e1 = __expf(wf1 - wm);
    float nw0 = e0 / (e0 + e1), nw1 = e1 / (e0 + e1);
    // ---- masked softmax over j for each path, scale by fuse weight ----
#pragma unroll
    for (int path = 0; path < 2; ++path) {
      v8f* cc = path ? cf : ct;
      float scale = path ? nw1 : nw0;
      float mx = -1e30f;
#pragma unroll
      for (int mt = 0; mt < 3; ++mt)
#pragma unroll
        for (int v = 0; v < 8; ++v) {
          int j = mt * 16 + v + 8 * hi;
          if (j < NTOP) { float x = cc[mt][v] * SC; mx = fmaxf(mx, x); }
        }
      mx = fmaxf(mx, __shfl_xor(mx, 16));
      float sum = 0.f;
#pragma unroll
      for (int mt = 0; mt < 3; ++mt)
#pragma unroll
        for (int v = 0; v < 8; ++v) {
          int j = mt * 16 + v + 8 * hi;
          float p = 0.f;
          if (j < NTOP) { p = __expf(cc[mt][v] * SC - mx); sum += p; }
          cc[mt][v] = p;
        }
      sum += __shfl_xor(sum, 16);
      float rs = scale / sum;
#pragma unroll
      for (int mt = 0; mt < 3; ++mt)
#pragma unroll
        for (int v = 0; v < 8; ++v) {
          int j = path * JPAD + mt * 16 + v + 8 * hi;
          Pw[j * 16 + n] = cc[mt][v] * rs;
        }
    }
    __syncthreads();   // uniform: every wave iterates st-loop exactly 8 times
    // ---- P.V GEMM:  C[m=s, n=d], K = j (96) ----
    v8f co[4] = {};
    for (int kk = 0; kk < 24; ++kk) {
      int j0 = kk * 4 + hi * 2;
      v2f a; a.x = Pw[j0 * 16 + n]; a.y = Pw[(j0 + 1) * 16 + n];
      size_t va0 = bhq + (size_t)jdx[j0] * 512;
      size_t va1 = bhq + (size_t)jdx[j0 + 1] * 512;
#pragma unroll
      for (int nt = 0; nt < 4; ++nt) {
        v2f bb;
        bb.x = vg[va0 + nt * 16 + n];
        bb.y = vg[va1 + nt * 16 + n];
        co[nt] = wmma_f32(a, bb, co[nt]);
      }
    }
    size_t ob = ((size_t)(b * 8 + h) * 64) * 1024;
#pragma unroll
    for (int nt = 0; nt < 4; ++nt)
#pragma unroll
      for (int v = 0; v < 8; ++v) {
        int d = nt * 16 + n;
        int s2 = s0 + v + 8 * hi;
        out[ob + (size_t)d * 1024 + s2] = 0.5f * co[nt][v];
      }
  }
}

// =========================================================================
// Kernel 5: forward DFT (64 modes) of q and k:  X[e, 0:64|64:128] = re|im.
// grid: 64 blocks = {q,k} x (b,h); GEMM M=e(64) N=128 K=l(1024), WMMA.
// =========================================================================
__global__ __launch_bounds__(256) void k_fwd_dft(const float* __restrict__ qg,
                                                 const float* __restrict__ kg,
                                                 const float* __restrict__ dftF,
                                                 float* __restrict__ XQ,
                                                 float* __restrict__ XK) {
  int blk = blockIdx.x;
  int sel = blk >> 5;
  int bh = blk & 31;
  int b = bh >> 3, h = bh & 7;
  const float* src = sel ? kg : qg;
  float* dst = (sel ? XK : XQ) + (size_t)bh * 64 * 128;
  int t = threadIdx.x, wid = t >> 5, lane = t & 31, n = lane & 15, hi = lane >> 4;
  int mt = wid & 3, nh = wid >> 2;
  size_t base = qidx(b, 0, h, 0);
  v8f acc[4] = {};
  for (int kk = 0; kk < 256; ++kk) {
    int l = kk * 4 + hi * 2;
    size_t a0 = base + (size_t)l * 512 + mt * 16 + n;
    if ((kk & 15) == 0) __builtin_prefetch(&src[a0 + 4 * 512], 0, 1);
    v2f a; a.x = src[a0]; a.y = src[a0 + 512];
#pragma unroll
    for (int q = 0; q < 4; ++q) {
      int nt = nh * 4 + q;
      v2f bb;
      bb.x = dftF[l * 128 + nt * 16 + n];
      bb.y = dftF[(l + 1) * 128 + nt * 16 + n];
      acc[q] = wmma_f32(a, bb, acc[q]);
    }
  }
#pragma unroll
  for (int q = 0; q < 4; ++q) {
    int nt = nh * 4 + q;
#pragma unroll
    for (int v = 0; v < 8; ++v) {
      int e = mt * 16 + v + 8 * hi;
      dst[e * 128 + nt * 16 + n] = acc[q][v];
    }
  }
}

// =========================================================================
// Kernel 6: xqk = ctanh( XQ^T * XK )  (complex 64x64 per (b,h)), WMMA.
//   Pre = Qr^T Kr - Qi^T Ki ; Pim = Qr^T Ki + Qi^T Kr ;
//   ctanh(a+bi) = (sinh 2a + i sin 2b) / (cosh 2a + cos 2b), clamped.
// =========================================================================
__global__ __launch_bounds__(256) void k_xqk(const float* __restrict__ XQ,
                                             const float* __restrict__ XK,
                                             float* __restrict__ Pre,
                                             float* __restrict__ Pim) {
  int bh = blockIdx.x;
  const float* Q = XQ + (size_t)bh * 8192;
  const float* K = XK + (size_t)bh * 8192;
  float* pr = Pre + (size_t)bh * 4096;
  float* pi = Pim + (size_t)bh * 4096;
  int t = threadIdx.x, wid = t >> 5, lane = t & 31, n = lane & 15, hi = lane >> 4;
  for (int blkI = wid; blkI < 16; blkI += 8) {
    int xt = blkI >> 2, yt = blkI & 3;
    v8f cre = {}, cim = {};
    for (int kk = 0; kk < 16; ++kk) {
      int e = kk * 4 + hi * 2;
      v2f aqr, aqi, bkr, bki, bkiN;
      aqr.x = Q[e * 128 + xt * 16 + n];      aqr.y = Q[(e + 1) * 128 + xt * 16 + n];
      aqi.x = Q[e * 128 + 64 + xt * 16 + n]; aqi.y = Q[(e + 1) * 128 + 64 + xt * 16 + n];
      bkr.x = K[e * 128 + yt * 16 + n];      bkr.y = K[(e + 1) * 128 + yt * 16 + n];
      bki.x = K[e * 128 + 64 + yt * 16 + n]; bki.y = K[(e + 1) * 128 + 64 + yt * 16 + n];
      bkiN.x = -bki.x; bkiN.y = -bki.y;
      cre = wmma_f32(aqr, bkr, cre);
      cre = wmma_f32(aqi, bkiN, cre);
      cim = wmma_f32(aqr, bki, cim);
      cim = wmma_f32(aqi, bkr, cim);
    }
#pragma unroll
    for (int v = 0; v < 8; ++v) {
      float a2 = fminf(fmaxf(2.0f * cre[v], -30.0f), 30.0f);
      float s2b, c2b;
      __sincosf(2.0f * cim[v], &s2b, &c2b);
      float den = coshf(a2) + c2b;
      int x = xt * 16 + v + 8 * hi;
      int y = yt * 16 + n;
      pr[x * 64 + y] = sinhf(a2) / den;
      pi[x * 64 + y] = s2b / den;
    }
  }
}

// =========================================================================
// Kernel 7: xqkv[e,x] = sum_y xqk[x,y] * xk_ft[e,y]  (complex), WMMA.
// =========================================================================
__global__ __launch_bounds__(256) void k_xqkv(const float* __restrict__ XK,
                                              const float* __restrict__ Pre,
                                              const float* __restrict__ Pim,
                                              float* __restrict__ XV) {
  int bh = blockIdx.x;
  const float* K = XK + (size_t)bh * 8192;
  const float* pr = Pre + (size_t)bh * 4096;
  const float* pi = Pim + (size_t)bh * 4096;
  float* V = XV + (size_t)bh * 8192;
  int t = threadIdx.x, wid = t >> 5, lane = t & 31, n = lane & 15, hi = lane >> 4;
  for (int blkI = wid; blkI < 16; blkI += 8) {
    int et = blkI >> 2, xt = blkI & 3;
    v8f cre = {}, cim = {};
    for (int kk = 0; kk < 16; ++kk) {
      int y = kk * 4 + hi * 2;
      int e = et * 16 + n;
      v2f akr, aki, bpr, bpi, bpiN;
      akr.x = K[e * 128 + y];      akr.y = K[e * 128 + y + 1];
      aki.x = K[e * 128 + 64 + y]; aki.y = K[e * 128 + 64 + y + 1];
      int x = xt * 16 + n;
      bpr.x = pr[x * 64 + y]; bpr.y = pr[x * 64 + y + 1];
      bpi.x = pi[x * 64 + y]; bpi.y = pi[x * 64 + y + 1];
      bpiN.x = -bpi.x; bpiN.y = -bpi.y;
      cre = wmma_f32(akr, bpr, cre);   // + Kr*Pre
      cre = wmma_f32(aki, bpiN, cre);  // - Ki*Pim
      cim = wmma_f32(aki, bpr, cim);   // + Ki*Pre
      cim = wmma_f32(akr, bpi, cim);   // + Kr*Pim
    }
#pragma unroll
    for (int v = 0; v < 8; ++v) {
      int e = et * 16 + v + 8 * hi;
      int x = xt * 16 + n;
      V[e * 128 + x] = cre[v];
      V[e * 128 + 64 + x] = cim[v];
    }
  }
}

// =========================================================================
// Kernel 8: per-mode weight contraction (complex, small: 33 M MACs, VALU):
//   XW[b,h,o,x] = sum_e w1[h,e,o,x] * XV[b,h,e,x]
// =========================================================================
__global__ void k_w1(const float* __restrict__ w1re, const float* __restrict__ w1im,
                     const float* __restrict__ XV, float* __restrict__ XW) {
  int t = blockIdx.x * blockDim.x + threadIdx.x;
  if (t >= BB * HH * 64 * 64) return;
  int x = t & 63, o = (t >> 6) & 63, h = (t >> 12) & 7, b = t >> 15;
  const float* V = XV + (size_t)(b * 8 + h) * 8192;
  float ar = 0.f, ai = 0.f;
  for (int e = 0; e < 64; ++e) {
    float wr = w1re[(((h * 64 + e) * 64 + o) * 64) + x];
    float wi = w1im[(((h * 64 + e) * 64 + o) * 64) + x];
    float vr = V[e * 128 + x];
    float vi = V[e * 128 + 64 + x];
    ar += wr * vr - wi * vi;
    ai += wr * vi + wi * vr;
  }
  float* W = XW + (size_t)(b * 8 + h) * 8192;
  W[o * 128 + x] = ar;
  W[o * 128 + 64 + x] = ai;
}

// =========================================================================
// Kernel 9: inverse DFT (WMMA): d_out += XW (o x 128) @ dftI (128 x 1024).
// Runs after k_attn, so "+=" is replay-safe.
// =========================================================================
__global__ __launch_bounds__(256) void k_inv_dft(const float* __restrict__ XW,
                                                 const float* __restrict__ dftI,
                                                 float* __restrict__ out) {
  int bh = blockIdx.x;
  const float* W = XW + (size_t)bh * 8192;
  float* ob = out + (size_t)bh * 64 * 1024;
  int t = threadIdx.x, wid = t >> 5, lane = t & 31, n = lane & 15, hi = lane >> 4;
  int mt = wid & 3, nhf = wid >> 2;
  for (int q = 0; q < 32; ++q) {
    int nt = nhf * 32 + q;
    v8f c = {};
    for (int kk = 0; kk < 32; ++kk) {
      int k0 = kk * 4 + hi * 2;
      int o = mt * 16 + n;
      v2f a; a.x = W[o * 128 + k0]; a.y = W[o * 128 + k0 + 1];
      v2f bb;
      bb.x = dftI[k0 * 1024 + nt * 16 + n];
      bb.y = dftI[(k0 + 1) * 1024 + nt * 16 + n];
      c = wmma_f32(a, bb, c);
    }
#pragma unroll
    for (int v = 0; v < 8; ++v) {
      int o = mt * 16 + v + 8 * hi;
      int l = nt * 16 + n;
      ob[(size_t)o * 1024 + l] += c[v];
    }
  }
}

// =========================================================================
extern "C" void kernel_launch(void* const* d_in, const int* in_sizes, int n_in,
                              void* d_out, int out_size, void* d_ws, size_t ws_size,
                              hipStream_t stream) {
  const float* tfq   = (const float*)d_in[0];
  const float* qg    = (const float*)d_in[1];
  const float* kg    = (const float*)d_in[2];
  const float* vg    = (const float*)d_in[3];
  // d_in[4] = mask, unused by the reference
  const float* wfuse = (const float*)d_in[5];
  const float* w1re  = (const float*)d_in[6];
  const float* w1im  = (const float*)d_in[7];
  float* out = (float*)d_out;

  char* ws = (char*)d_ws;
  size_t off = 0;
  auto alloc = [&](size_t nfl) { float* p = (float*)(ws + off); off += nfl * sizeof(float); return p; };
  float* dftF  = alloc(1024 * 128);
  float* dftI  = alloc(128 * 1024);
  float* ampT  = alloc(32 * 1024);
  float* ampTF = alloc(32 * 1024);
  int*   idxT  = (int*)alloc(32 * 64);
  int*   idxTF = (int*)alloc(32 * 64);
  float* XQ    = alloc(32 * 8192);
  float* XK    = alloc(32 * 8192);
  float* Pre   = alloc(32 * 4096);
  float* Pim   = alloc(32 * 4096);
  float* XV    = alloc(32 * 8192);
  float* XW    = alloc(32 * 8192);
  (void)ws_size; (void)in_sizes; (void)n_in; (void)out_size;

  hipMemsetAsync(ampT, 0, 32 * 1024 * sizeof(float), stream);

  k_build_dft<<<256, 256, 0, stream>>>(dftF, dftI);
  k_corr_amp<<<2048, 256, 0, stream>>>(qg, kg, ampT);
  k_tf_amp<<<128, 256, 0, stream>>>(tfq, ampTF);
  k_topk<<<64, 256, 0, stream>>>(ampT, ampTF, idxT, idxTF);
  k_attn<<<32, 256, 0, stream>>>(qg, tfq, kg, vg, wfuse, idxT, idxTF, out);
  k_fwd_dft<<<64, 256, 0, stream>>>(qg, kg, dftF, XQ, XK);
  k_xqk<<<32, 256, 0, stream>>>(XQ, XK, Pre, Pim);
  k_xqkv<<<32, 256, 0, stream>>>(XK, Pre, Pim, XV);
  k_w1<<<512, 256, 0, stream>>>(w1re, w1im, XV, XW);
  k_inv_dft<<<32, 256, 0, stream>>>(XW, dftI, out);
}